// CholeskyResHead_30434138259538
// MI455X (gfx1250) — compile-verified
//
#include <hip/hip_runtime.h>
#include <math.h>

// ---------------------------------------------------------------------------
// CholeskyResHead loss for MI455X (gfx1250, wave32, WMMA f32 16x16x4)
// B=2048, N=207, T=12, R=4
// ---------------------------------------------------------------------------

#define Bn   2048
#define Nn   207
#define Tn   12
#define Rn   4
#define NP   208            // padded spatial dim
#define KS   (NP/2)         // 104 k-step pairs
#define NT   (Nn*Tn)        // 2484

typedef __attribute__((ext_vector_type(2))) float v2f;
typedef __attribute__((ext_vector_type(8))) float v8f;

// ---------------- one-sided Jacobi pair rotation (8 lanes per pair) ---------
__device__ __forceinline__ void jacobi_pair(float* colp, float* colq, int rows, int sub) {
  float app = 0.f, bqq = 0.f, cpq = 0.f;
  for (int rI = sub; rI < rows; rI += 8) {
    float wp = colp[rI], wq = colq[rI];
    app += wp * wp; bqq += wq * wq; cpq += wp * wq;
  }
  #pragma unroll
  for (int d = 4; d > 0; d >>= 1) {
    app += __shfl_xor(app, d, 8);
    bqq += __shfl_xor(bqq, d, 8);
    cpq += __shfl_xor(cpq, d, 8);
  }
  if (cpq != 0.0f && cpq * cpq > 1e-26f * app * bqq) {
    float zeta = (bqq - app) / (2.0f * cpq);
    float t    = ((zeta >= 0.f) ? 1.0f : -1.0f) / (fabsf(zeta) + sqrtf(1.0f + zeta * zeta));
    float cs   = 1.0f / sqrtf(1.0f + t * t);
    float sn   = cs * t;
    for (int rI = sub; rI < rows; rI += 8) {
      float wp = colp[rI], wq = colq[rI];
      colp[rI] = cs * wp - sn * wq;
      colq[rI] = sn * wp + cs * wq;
    }
  }
}

// ---------------- spatial eigensolver: one-sided Jacobi SVD of L+eps*I ------
// Output UsA in WMMA-A-fragment layout: float2[(r*KS + ks)*NP + i] =
//   ( U[n=2ks][i], U[n=2ks+1][i] )   -- one b64 load per A fragment.
// grid = R blocks, 832 threads (104 pairs x 8), dynamic LDS = (208*208+256)*4 B
__global__ __launch_bounds__(832)
void eig_spatial_kernel(const float* __restrict__ Ls, float* __restrict__ UsA,
                        float* __restrict__ Ds, float* __restrict__ scal) {
  extern __shared__ float sm[];
  float* W    = sm;               // column-major: W[c*NP + row]
  float* invs = sm + NP * NP;     // 208 (+pad)
  const int r   = blockIdx.x;
  const int tid = threadIdx.x;
  const float* L = Ls + r * Nn * Nn;

  for (int e = tid; e < NP * NP; e += 832) {
    int c = e / NP, rw = e % NP;
    float v = 0.f;
    if (c < Nn && rw < Nn) { v = L[rw * Nn + c]; if (rw == c) v += 1e-6f; }
    W[c * NP + rw] = v;
  }
  __syncthreads();

  const int pairI = tid >> 3, sub = tid & 7;
  for (int sweep = 0; sweep < 14; ++sweep) {
    for (int rr = 0; rr < NP - 1; ++rr) {       // 207 rounds, tournament schedule
      int p, q;
      if (pairI == 0) { p = NP - 1; q = rr; }
      else { p = (rr + pairI) % (NP - 1); q = (rr - pairI + (NP - 1)) % (NP - 1); }
      jacobi_pair(&W[p * NP], &W[q * NP], NP, sub);
      __syncthreads();
    }
  }
  if (tid < NP) {                               // D = S^2, invs = 1/S
    float s2 = 0.f;
    const float* col = &W[tid * NP];
    for (int rw = 0; rw < NP; ++rw) s2 += col[rw] * col[rw];
    Ds[r * NP + tid] = s2;
    invs[tid] = (s2 > 0.f) ? (1.0f / sqrtf(s2)) : 0.f;
  }
  __syncthreads();
  for (int e = tid; e < NP * NP; e += 832) {    // store in A-fragment layout
    int i = e % NP, n = e / NP;
    float v = W[i * NP + n] * invs[i];
    int ks = n >> 1, par = n & 1;
    UsA[(((r * KS + ks) * NP) + i) * 2 + par] = v;
  }
  if (tid == 0) {
    float ld = 0.f;
    for (int d = 0; d < Nn; ++d) ld += logf(L[d * Nn + d]);
    scal[r * 3 + 0] = ld;                       // Ulogdet (spatial)
  }
}

// ---------------- temporal eigensolver (12x12) + scalars --------------------
__global__ __launch_bounds__(64)
void eig_temporal_kernel(const float* __restrict__ Lt, const float* __restrict__ sigma,
                         float* __restrict__ Ut, float* __restrict__ Dt,
                         float* __restrict__ scal) {
  __shared__ float W[Tn * Tn];
  __shared__ float invs[Tn];
  const int r = blockIdx.x, tid = threadIdx.x;
  const float* L = Lt + r * Tn * Tn;
  for (int e = tid; e < Tn * Tn; e += 64) {
    int c = e / Tn, rw = e % Tn;
    float v = L[rw * Tn + c];
    if (rw == c) v += 1e-6f;
    W[c * Tn + rw] = v;
  }
  __syncthreads();
  const int pairI = tid >> 3, sub = tid & 7;
  for (int sweep = 0; sweep < 12; ++sweep) {
    for (int rr = 0; rr < Tn - 1; ++rr) {
      if (pairI < Tn / 2) {
        int p, q;
        if (pairI == 0) { p = Tn - 1; q = rr; }
        else { p = (rr + pairI) % (Tn - 1); q = (rr - pairI + (Tn - 1)) % (Tn - 1); }
        jacobi_pair(&W[p * Tn], &W[q * Tn], Tn, sub);
      }
      __syncthreads();
    }
  }
  if (tid < Tn) {
    float s2 = 0.f;
    for (int rw = 0; rw < Tn; ++rw) { float v = W[tid * Tn + rw]; s2 += v * v; }
    Dt[r * 16 + tid] = s2;
    invs[tid] = (s2 > 0.f) ? (1.0f / sqrtf(s2)) : 0.f;
  }
  __syncthreads();
  for (int e = tid; e < 256; e += 64) {         // U_t padded 16x16: Ut[r][t][l]
    int l = e & 15, t = e >> 4;
    float v = 0.f;
    if (l < Tn && t < Tn) v = W[l * Tn + t] * invs[l];
    Ut[r * 256 + t * 16 + l] = v;
  }
  if (tid == 0) {
    float ld = 0.f;
    for (int d = 0; d < Tn; ++d) ld += logf(L[d * Tn + d]);
    scal[r * 3 + 1] = ld;                       // Vlogdet (temporal)
    float s = 0.1f / (1.0f + expf(-sigma[r]));
    scal[r * 3 + 2] = s * s;                    // sig^2
  }
}

// ---------------- main per-batch kernel: WMMA GEMM + mah + mse --------------
// grid = B blocks, 256 threads (8 waves)
__global__ __launch_bounds__(256)
void main_kernel(const float* __restrict__ target, const float* __restrict__ utg,
                 const float* __restrict__ mu, const float* __restrict__ w,
                 const float* __restrict__ UsA, const float* __restrict__ Ds,
                 const float* __restrict__ Ut, const float* __restrict__ Dt,
                 const float* __restrict__ scal,
                 float* __restrict__ nll_b, float* __restrict__ sq_b,
                 float* __restrict__ cnt_b) {
  extern __shared__ float sm[];
  float* res  = sm;                       // [4][208][12]
  float* resu = res  + Rn * NP * Tn;      // paired: float2[(r*KS+ks)*16 + l]
  float* utl  = resu + Rn * NP * 16;      // [4][16][16]
  float* mah  = utl  + Rn * 256;          // 64 (52 tile partials)
  float* red  = mah  + 64;                // 256
  float* red2 = red  + 256;               // 256

  const int b = blockIdx.x, tid = threadIdx.x;
  const float ew0 = expf(w[b * 4 + 0]), ew1 = expf(w[b * 4 + 1]);
  const float ew2 = expf(w[b * 4 + 2]), ew3 = expf(w[b * 4 + 3]);
  const float*  tg = target + b * NT;
  const float*  up = utg    + b * NT;
  const float4* mp = (const float4*)(mu + (size_t)b * NT * 4);

  // ---- stage 1: residuals to LDS + masked-MSE partials ----
  float sqacc = 0.f, cntacc = 0.f;
  for (int e = tid; e < NT; e += 256) {
    float t  = tg[e];
    float4 m = mp[e];                           // r fastest -> coalesced 128b
    int n = e / Tn, tt = e - n * Tn;
    res[(0 * NP + n) * Tn + tt] = m.x - t;
    res[(1 * NP + n) * Tn + tt] = m.y - t;
    res[(2 * NP + n) * Tn + tt] = m.z - t;
    res[(3 * NP + n) * Tn + tt] = m.w - t;
    float pred = m.x * ew0 + m.y * ew1 + m.z * ew2 + m.w * ew3;
    float ind  = (up[e] != 0.0f) ? 1.f : 0.f;
    float d    = pred - t;
    sqacc  += ind * d * d;
    cntacc += ind;
  }
  for (int e = tid; e < Rn * 256; e += 256) utl[e] = Ut[e];
  __syncthreads();

  // ---- stage 2: ResU[n][l] = sum_t Res[n][t]*U_t[t][l], paired-row layout --
  for (int o = tid; o < Rn * NP * 16; o += 256) {
    int r = o / (NP * 16), rem = o - r * (NP * 16);
    int n = rem >> 4, l = rem & 15;
    float acc = 0.f;
    if (n < Nn && l < Tn) {
      const float* rp = &res[(r * NP + n) * Tn];
      const float* uq = &utl[r * 256 + l];
      #pragma unroll
      for (int t = 0; t < Tn; ++t) acc += rp[t] * uq[t * 16];
    }
    resu[((((r * KS) + (n >> 1)) * 16 + l) << 1) | (n & 1)] = acc;
  }
  __syncthreads();

  // ---- stage 3: KV = U_s^T @ ResU via v_wmma_f32_16x16x4_f32 ----
  // 28 dual-tile tasks: r in 0..3, pair pi in 0..6 covering tiles (pi, pi+7)
  const int wave = tid >> 5, lane = tid & 31;
  const int m = lane & 15, kh = lane >> 4;
  const v2f* resuF2 = (const v2f*)resu;
  const v2f* usF2   = (const v2f*)UsA;
  for (int task = wave; task < 28; task += 8) {
    const int r  = task >> 3 ? (task / 7) : (task / 7);   // task/7
    const int rT = task / 7, pi = task - rT * 7;
    const int t0 = pi;
    const bool two = (pi < 6);
    const int t1 = two ? pi + 7 : 12;                     // clamped; epilogue skipped
    const v2f* ap0 = usF2 + (size_t)(rT * KS + kh) * NP + t0 * 16 + m;
    const v2f* ap1 = usF2 + (size_t)(rT * KS + kh) * NP + t1 * 16 + m;
    const v2f* bp  = resuF2 + (rT * KS + kh) * 16 + m;
    v8f acc0 = {0.f,0.f,0.f,0.f,0.f,0.f,0.f,0.f};
    v8f acc1 = {0.f,0.f,0.f,0.f,0.f,0.f,0.f,0.f};
    for (int kb = 0; kb < KS / 2; kb += 4) {              // 52 kb, unroll 4
      #pragma unroll
      for (int u = 0; u < 4; ++u) {
        v2f bv = bp[(size_t)(kb + u) * 32];               // ds_load_b64
        v2f a0 = ap0[(size_t)(kb + u) * 2 * NP];          // global_load_b64
        v2f a1 = ap1[(size_t)(kb + u) * 2 * NP];
        acc0 = __builtin_amdgcn_wmma_f32_16x16x4_f32(false, a0, false, bv,
                                                     (short)0, acc0, false, false);
        acc1 = __builtin_amdgcn_wmma_f32_16x16x4_f32(false, a1, false, bv,
                                                     (short)0, acc1, false, false);
      }
    }
    // weight by 1/(D_s[i]*D_t[l] + sig^2), square-sum, wave-reduce
    const float sig2 = scal[rT * 3 + 2];
    const float dtl  = (m < Tn) ? Dt[rT * 16 + m] : 0.f;
    float p0 = 0.f, p1 = 0.f;
    #pragma unroll
    for (int v = 0; v < 8; ++v) {
      int i0 = t0 * 16 + v + kh * 8;
      int i1 = t1 * 16 + v + kh * 8;
      if (i0 < Nn && m < Tn) {
        float val = acc0[v];
        p0 += (val * val) / (Ds[rT * NP + i0] * dtl + sig2);
      }
      if (two && i1 < Nn && m < Tn) {
        float val = acc1[v];
        p1 += (val * val) / (Ds[rT * NP + i1] * dtl + sig2);
      }
    }
    #pragma unroll
    for (int d = 16; d > 0; d >>= 1) {
      p0 += __shfl_xor(p0, d, 32);
      p1 += __shfl_xor(p1, d, 32);
    }
    if (lane == 0) {
      mah[rT * 13 + t0] = p0;
      if (two) mah[rT * 13 + t1] = p1;
    }
  }
  __syncthreads();

  // ---- block reductions (fixed order, deterministic) ----
  red[tid] = sqacc; red2[tid] = cntacc;
  __syncthreads();
  for (int s = 128; s > 0; s >>= 1) {
    if (tid < s) { red[tid] += red[tid + s]; red2[tid] += red2[tid + s]; }
    __syncthreads();
  }
  if (tid == 0) {
    sq_b[b]  = red[0];
    cnt_b[b] = red2[0];
    const float C0 = (float)(-0.5 * (double)NT * 1.8378770664093453);  // -NT/2*ln(2pi)
    float terms[Rn], mx = -1e30f;
    #pragma unroll
    for (int r = 0; r < Rn; ++r) {
      float mh = 0.f;
      for (int mt = 0; mt < 13; ++mt) mh += mah[r * 13 + mt];
      float tv = C0 - 0.5f * mh + (float)Nn * scal[r * 3 + 1]
               + (float)Tn * scal[r * 3 + 0] + w[b * 4 + r];
      terms[r] = tv; mx = fmaxf(mx, tv);
    }
    float se = 0.f;
    #pragma unroll
    for (int r = 0; r < Rn; ++r) se += expf(terms[r] - mx);
    nll_b[b] = -(mx + logf(se));
  }
}

// ---------------- final reduction over B ------------------------------------
__global__ __launch_bounds__(256)
void finalize_kernel(const float* __restrict__ nll_b, const float* __restrict__ sq_b,
                     const float* __restrict__ cnt_b, float* __restrict__ out) {
  __shared__ float r0[256], r1[256], r2[256];
  const int tid = threadIdx.x;
  float a = 0.f, s = 0.f, c = 0.f;
  for (int i = tid; i < Bn; i += 256) { a += nll_b[i]; s += sq_b[i]; c += cnt_b[i]; }
  r0[tid] = a; r1[tid] = s; r2[tid] = c;
  __syncthreads();
  for (int st = 128; st > 0; st >>= 1) {
    if (tid < st) { r0[tid] += r0[tid + st]; r1[tid] += r1[tid + st]; r2[tid] += r2[tid + st]; }
    __syncthreads();
  }
  if (tid == 0) {
    float nll = r0[0] / (float)Bn;
    float mse = (r2[0] > 0.f) ? (r1[0] / r2[0]) : 0.f;
    out[0] = 0.1f * nll + 0.9f * mse;
    out[1] = nll;
    out[2] = mse;
  }
}

// ---------------------------------------------------------------------------
extern "C" void kernel_launch(void* const* d_in, const int* in_sizes, int n_in,
                              void* d_out, int out_size, void* d_ws, size_t ws_size,
                              hipStream_t stream) {
  (void)in_sizes; (void)n_in; (void)out_size; (void)ws_size;
  const float* target = (const float*)d_in[0];
  const float* utg    = (const float*)d_in[1];
  const float* mu     = (const float*)d_in[2];
  const float* w      = (const float*)d_in[3];
  const float* sigma  = (const float*)d_in[4];
  const float* Ls     = (const float*)d_in[5];
  const float* Lt     = (const float*)d_in[6];

  float* ws    = (float*)d_ws;
  float* UsA   = ws;                    // 4*104*208*2 = 173,056
  float* Ds    = UsA + Rn * KS * NP * 2;
  float* Ut    = Ds + Rn * NP;          // 4*256
  float* Dt    = Ut + Rn * 256;         // 4*16
  float* scal  = Dt + Rn * 16;          // 16
  float* nllb  = scal + 16;             // 2048
  float* sqb   = nllb + Bn;             // 2048
  float* cntb  = sqb + Bn;              // 2048

  size_t smem_eig  = (size_t)(NP * NP + 256) * sizeof(float);                    // ~174 KB
  size_t smem_main = (size_t)(Rn * NP * Tn + Rn * NP * 16 + Rn * 256 + 64 + 512) * sizeof(float); // ~100 KB

  eig_spatial_kernel<<<Rn, 832, smem_eig, stream>>>(Ls, UsA, Ds, scal);
  eig_temporal_kernel<<<Rn, 64, 0, stream>>>(Lt, sigma, Ut, Dt, scal);
  main_kernel<<<Bn, 256, smem_main, stream>>>(target, utg, mu, w, UsA, Ds, Ut, Dt,
                                              scal, nllb, sqb, cntb);
  finalize_kernel<<<1, 256, 0, stream>>>(nllb, sqb, cntb, (float*)d_out);
}